// Encoder_P_66675072303712
// MI455X (gfx1250) — compile-verified
//
#include <hip/hip_runtime.h>

#define HH 512
#define WW 512
#define BB 8
#define PP (HH * WW)          // 262144 plane size
#define CHO 48
#define PI_F 3.14159265358979323846f
#define TWOPI_F 6.28318530717958647692f

typedef __attribute__((ext_vector_type(2))) float v2f;
typedef __attribute__((ext_vector_type(8))) float v8f;

__device__ __forceinline__ float wrap_diff(float d) {
    // f(d) = |d|<pi ? d : (mod(d+pi, 2pi) - pi, with (-pi & d>0) -> +pi)
    float t = d + PI_F;
    float m = t - floorf(t * (1.0f / TWOPI_F)) * TWOPI_F;   // floored mod -> [0, 2pi)
    float dd = m - PI_F;
    if (dd == -PI_F && d > 0.0f) dd = PI_F;
    return (fabsf(d) < PI_F) ? d : dd;
}

// ---------------- Kernel 1: sq = (diff of unwrap, padded)^2, broadcast to 8 output channels
__global__ __launch_bounds__(256) void k_sq(const float* __restrict__ p, float* __restrict__ out) {
    int idx = blockIdx.x * blockDim.x + threadIdx.x;   // B * H * (W/4)
    int b   = idx >> 16;                               // / (512*128)
    int rem = idx & 0xFFFF;
    int h   = rem >> 7;
    int x   = (rem & 127) << 2;

    const float* rowc = p + b * PP + h * WW + x;
    float4 v;
    if (h == 0) {
        v = make_float4(0.f, 0.f, 0.f, 0.f);
    } else {
        float4 cur = *(const float4*)rowc;
        float4 prv = *(const float4*)(rowc - WW);
        float d0 = wrap_diff(cur.x - prv.x);
        float d1 = wrap_diff(cur.y - prv.y);
        float d2 = wrap_diff(cur.z - prv.z);
        float d3 = wrap_diff(cur.w - prv.w);
        v = make_float4(d0 * d0, d1 * d1, d2 * d2, d3 * d3);
    }
    float* ob = out + b * CHO * PP + h * WW + x;
    const int chans[8] = {30, 31, 34, 35, 42, 43, 46, 47};
#pragma unroll
    for (int i = 0; i < 8; ++i)
        *(float4*)(ob + chans[i] * PP) = v;
}

// ---------------- Kernel 2: conv1  1->2, 4x4, dil 1, pad (top1,bot2,left1,right2)
__global__ __launch_bounds__(256) void k_conv1(const float* __restrict__ w1, const float* __restrict__ b1,
                                               float* __restrict__ out) {
    __shared__ float sw[32];
    __shared__ float sb[2];
    int tid = threadIdx.x;
    if (tid < 32) sw[tid] = w1[tid];
    if (tid < 2)  sb[tid] = b1[tid];
    __syncthreads();

    int idx = blockIdx.x * blockDim.x + tid;           // B*H*W
    int b   = idx >> 18;
    int rem = idx & (PP - 1);
    int y   = rem >> 9;
    int x   = rem & 511;

    const float* sq = out + b * CHO * PP + 47 * PP;
    float a0 = sb[0], a1 = sb[1];
#pragma unroll
    for (int ky = 0; ky < 4; ++ky) {
        int yy = y + ky - 1;
        if (yy < 0 || yy >= HH) continue;
#pragma unroll
        for (int kx = 0; kx < 4; ++kx) {
            int xx = x + kx - 1;
            if (xx < 0 || xx >= WW) continue;
            float s = sq[yy * WW + xx];
            a0 += s * sw[ky * 4 + kx];
            a1 += s * sw[16 + ky * 4 + kx];
        }
    }
    float* ob = out + b * CHO * PP + y * WW + x;
    ob[28 * PP] = a0; ob[32 * PP] = a0; ob[40 * PP] = a0; ob[44 * PP] = a0;
    ob[29 * PP] = a1; ob[33 * PP] = a1; ob[41 * PP] = a1; ob[45 * PP] = a1;
}

// ---------------- Kernel 3: conv2  3->4, 3x3, dil 2, pad 2 each side. In = ch {44,45,47}
__global__ __launch_bounds__(256) void k_conv2(const float* __restrict__ w2, const float* __restrict__ b2,
                                               float* __restrict__ out) {
    __shared__ float sw[108];
    __shared__ float sb[4];
    int tid = threadIdx.x;
    if (tid < 108) sw[tid] = w2[tid];
    if (tid < 4)   sb[tid] = b2[tid];
    __syncthreads();

    int idx = blockIdx.x * blockDim.x + tid;
    int b   = idx >> 18;
    int rem = idx & (PP - 1);
    int y   = rem >> 9;
    int x   = rem & 511;

    const float* base = out + b * CHO * PP;
    const int inch[3] = {44, 45, 47};
    float acc[4] = {sb[0], sb[1], sb[2], sb[3]};
#pragma unroll
    for (int c = 0; c < 3; ++c) {
        const float* ip = base + inch[c] * PP;
#pragma unroll
        for (int ky = 0; ky < 3; ++ky) {
            int yy = y + 2 * (ky - 1);
            if (yy < 0 || yy >= HH) continue;
#pragma unroll
            for (int kx = 0; kx < 3; ++kx) {
                int xx = x + 2 * (kx - 1);
                if (xx < 0 || xx >= WW) continue;
                float s = ip[yy * WW + xx];
#pragma unroll
                for (int o = 0; o < 4; ++o)
                    acc[o] += s * sw[((o * 3 + c) * 3 + ky) * 3 + kx];
            }
        }
    }
    float* ob = out + b * CHO * PP + y * WW + x;
#pragma unroll
    for (int o = 0; o < 4; ++o) {
        ob[(24 + o) * PP] = acc[o];
        ob[(36 + o) * PP] = acc[o];
    }
}

// ---------------- Kernel 4: conv3  8->8, 2x2, dil 3, pad (top/left 1, bot/right 2). In = ch 36..43
__global__ __launch_bounds__(256) void k_conv3(const float* __restrict__ w3, const float* __restrict__ b3,
                                               float* __restrict__ out) {
    __shared__ float sw[256];
    __shared__ float sb[8];
    int tid = threadIdx.x;
    sw[tid] = w3[tid];                 // exactly 256 weights, 256 threads
    if (tid < 8) sb[tid] = b3[tid];
    __syncthreads();

    int idx = blockIdx.x * blockDim.x + tid;
    int b   = idx >> 18;
    int rem = idx & (PP - 1);
    int y   = rem >> 9;
    int x   = rem & 511;

    const float* base = out + b * CHO * PP + 36 * PP;   // 8 contiguous input planes
    float acc[8];
#pragma unroll
    for (int o = 0; o < 8; ++o) acc[o] = sb[o];

#pragma unroll
    for (int ky = 0; ky < 2; ++ky) {
        int yy = y + 3 * ky - 1;                        // offsets -1, +2
        if (yy < 0 || yy >= HH) continue;
#pragma unroll
        for (int kx = 0; kx < 2; ++kx) {
            int xx = x + 3 * kx - 1;
            if (xx < 0 || xx >= WW) continue;
#pragma unroll
            for (int c = 0; c < 8; ++c) {
                float s = base[c * PP + yy * WW + xx];
#pragma unroll
                for (int o = 0; o < 8; ++o)
                    acc[o] += s * sw[(o * 8 + c) * 4 + ky * 2 + kx];
            }
        }
    }
    float* ob = out + b * CHO * PP + y * WW + x;
#pragma unroll
    for (int o = 0; o < 8; ++o)
        ob[(16 + o) * PP] = acc[o];
}

// ---------------- Kernel 5: conv4  20->16, 1x1  ==  per-pixel GEMM via V_WMMA_F32_16X16X4_F32
// One wave per (batch, row). A = weights (M=16 x K=4 chunk, 5 chunks), reused across 32 x-tiles.
// In channels: k<8 -> 16+k ; k>=8 -> 28+k. Out channels 0..15.
__global__ __launch_bounds__(256) void k_conv4_wmma(const float* __restrict__ w4,
                                                    const float* __restrict__ b4,
                                                    float* __restrict__ out) {
    int gtid = blockIdx.x * blockDim.x + threadIdx.x;
    int wave = gtid >> 5;                  // 0 .. 4095
    int lane = threadIdx.x & 31;
    int b = wave >> 9;                     // / 512
    int y = wave & 511;
    int nlo = lane & 15;
    int hi  = lane >> 4;                   // 0: K pair {0,1} / rows 0-7 ; 1: K pair {2,3} / rows 8-15

    // A fragments: A(M=nlo, K) — lanes 0-15: K = 4q+{0,1}; lanes 16-31: K = 4q+{2,3}
    v2f a[5];
#pragma unroll
    for (int q = 0; q < 5; ++q) {
        int k0 = 4 * q + 2 * hi;
        a[q].x = w4[nlo * 20 + k0];
        a[q].y = w4[nlo * 20 + k0 + 1];
    }
    // C init with bias: VGPR r holds rows {r (lanes 0-15), r+8 (lanes 16-31)}
    v8f cinit;
#pragma unroll
    for (int r = 0; r < 8; ++r) cinit[r] = b4[r + 8 * hi];

    const float* ibase = out + b * CHO * PP + y * WW;
    float*       obase = out + b * CHO * PP + y * WW;

    for (int t = 0; t < 32; ++t) {
        int x0 = t * 16;
        v8f c = cinit;
#pragma unroll
        for (int q = 0; q < 5; ++q) {
            int k0  = 4 * q + 2 * hi;
            int ch0 = (k0 < 8) ? (16 + k0) : (28 + k0);   // k0,k0+1 never straddle 8
            v2f bf;
            bf.x = ibase[ch0 * PP + x0 + nlo];
            bf.y = ibase[(ch0 + 1) * PP + x0 + nlo];
            c = __builtin_amdgcn_wmma_f32_16x16x4_f32(
                    false, a[q], false, bf, (short)0, c, false, false);
        }
#pragma unroll
        for (int r = 0; r < 8; ++r)
            obase[(r + 8 * hi) * PP + x0 + nlo] = c[r];
    }
}

extern "C" void kernel_launch(void* const* d_in, const int* in_sizes, int n_in,
                              void* d_out, int out_size, void* d_ws, size_t ws_size,
                              hipStream_t stream) {
    const float* fin = (const float*)d_in[0];
    const float* w1  = (const float*)d_in[1];
    const float* b1  = (const float*)d_in[2];
    const float* w2  = (const float*)d_in[3];
    const float* b2  = (const float*)d_in[4];
    const float* w3  = (const float*)d_in[5];
    const float* b3  = (const float*)d_in[6];
    const float* w4  = (const float*)d_in[7];
    const float* b4  = (const float*)d_in[8];
    float* out = (float*)d_out;

    // All intermediates live in their final d_out channel slots; stream order gives the deps.
    k_sq       <<<(BB * HH * (WW / 4)) / 256, 256, 0, stream>>>(fin, out);
    k_conv1    <<<(BB * HH * WW) / 256,       256, 0, stream>>>(w1, b1, out);
    k_conv2    <<<(BB * HH * WW) / 256,       256, 0, stream>>>(w2, b2, out);
    k_conv3    <<<(BB * HH * WW) / 256,       256, 0, stream>>>(w3, b3, out);
    k_conv4_wmma<<<(BB * HH * 32) / 256,      256, 0, stream>>>(w4, b4, out);
}